// GeneticFitness_43413529428367
// MI455X (gfx1250) — compile-verified
//
#include <hip/hip_runtime.h>

typedef __attribute__((ext_vector_type(2))) float v2f;
typedef __attribute__((ext_vector_type(8))) float v8f;
typedef __attribute__((ext_vector_type(4))) float f4;

#define T_SEQ 2048
#define DDIM 64
#define WIN 256
#define GF_STRIDE 66     // 16x64 gene_fitness, padded (no bank clash; even -> 8B-aligned b64 reads)
#define BAND_STRIDE 290  // up to 272 band cols, padded so row m and m+8 hit disjoint bank sets

__global__ void __launch_bounds__(128, 4)
genetic_fitness_kernel(const float* __restrict__ pop, float* __restrict__ out)
{
    __shared__ float gf[16 * GF_STRIDE];     // inv-denom, then gene_fitness (in place)
    __shared__ float band[16 * BAND_STRIDE]; // masked org_fitness band
    __shared__ float sInv[16];               // 1 / sum_d inv_denom
    __shared__ float rowScale[16];           // 1 / row_sum

    const int tid = threadIdx.x;
    const int bh  = blockIdx.y;
    const int it  = blockIdx.x;      // i-tile index (16 rows)
    const int i0  = it * 16;
    const float* popBH = pop + (size_t)bh * (T_SEQ * DDIM);

    // ---- Phase 1: sliding-window sums -> inv_denom = 1/(mean + 0.5) ----
    {
        const int d    = tid & 63;
        const int half = tid >> 6;          // 0: rows 0..7, 1: rows 8..15
        const int r0   = i0 + half * 8;
        int lo = r0 - (WIN - 1); if (lo < 0) lo = 0;
        float s = 0.f;
        for (int j = lo; j <= r0; ++j) s += popBH[j * DDIM + d];
        #pragma unroll 1
        for (int m = 0; m < 8; ++m) {
            const int i = r0 + m;
            if (m) {
                s += popBH[i * DDIM + d];
                const int old = i - WIN;
                if (old >= 0) s -= popBH[old * DDIM + d];
            }
            const float cnt  = (float)((i + 1) < WIN ? (i + 1) : WIN);
            const float mean = s / cnt;
            gf[(half * 8 + m) * GF_STRIDE + d] = 1.0f / (mean + 0.5f);
        }
    }
    __syncthreads();

    // ---- Phase 2: per-row S = sum_d inv_denom; gene_fitness = inv_denom / S ----
    if (tid < 16) {
        float s = 0.f;
        for (int d = 0; d < DDIM; ++d) s += gf[tid * GF_STRIDE + d];
        sInv[tid] = 1.0f / s;
    }
    __syncthreads();
    for (int idx = tid; idx < 16 * DDIM; idx += 128) {
        const int m = idx >> 6, d = idx & 63;
        gf[m * GF_STRIDE + d] *= sInv[m];
    }
    __syncthreads();

    // ---- Phase 3: banded GEMM  org(16x16) = gf(16x64) x pop^T(64x16) via v_wmma_f32_16x16x4_f32 ----
    const int lane = tid & 31;
    const int wave = tid >> 5;
    const int n    = lane & 15;          // N (B/C/D column) for this lane
    const int kb   = (lane >> 4) * 2;    // K offset within 4-wide K chunk

    const int lowJ  = i0 - (WIN - 1);
    const int jt_lo = (lowJ <= 0) ? 0 : (lowJ >> 4);

    // A fragments: lane = M (16x4 f32 layout), loaded once from LDS, reused across all j-tiles
    v2f a[16];
    #pragma unroll
    for (int k = 0; k < 16; ++k) {
        const float* p = &gf[n * GF_STRIDE + 4 * k + kb];
        a[k].x = p[0];
        a[k].y = p[1];
    }

    // wave-uniform tile loop: EXEC stays all-ones around WMMA
    for (int jt = jt_lo + wave; jt <= it; jt += 4) {
        const int j0 = jt * 16;
        const float* rowp = popBH + (size_t)(j0 + n) * DDIM + kb;

        // batch all 16 B fragments first: loads issue as one clause,
        // latency overlaps instead of landing on the WMMA accumulation chain
        v2f b[16];
        #pragma unroll
        for (int k = 0; k < 16; ++k) {
            b[k].x = rowp[4 * k];
            b[k].y = rowp[4 * k + 1];
        }

        v8f c = {0.f, 0.f, 0.f, 0.f, 0.f, 0.f, 0.f, 0.f};
        #pragma unroll
        for (int k = 0; k < 16; ++k) {
            c = __builtin_amdgcn_wmma_f32_16x16x4_f32(
                    /*neg_a=*/false, a[k], /*neg_b=*/false, b[k],
                    /*c_mod=*/(short)0, c, /*reuse_a=*/false, /*reuse_b=*/false);
        }

        const int mBase = (lane >> 4) * 8;   // C/D layout: lanes 16-31 hold M = r + 8
        const int j = j0 + n;
        #pragma unroll
        for (int r = 0; r < 8; ++r) {
            const int i = i0 + mBase + r;
            const bool keep = (j <= i) && ((i - j) < WIN);
            band[(mBase + r) * BAND_STRIDE + (jt - jt_lo) * 16 + n] = keep ? c[r] : 0.0f;
        }
    }
    __syncthreads();

    // ---- Phase 4: row sums of masked band -> normalization scale ----
    if (tid < 16) {
        const int bw = (it - jt_lo + 1) * 16;
        float s = 0.f;
        for (int cix = 0; cix < bw; ++cix) s += band[tid * BAND_STRIDE + cix];
        rowScale[tid] = 1.0f / fmaxf(s, 1e-10f);
    }
    __syncthreads();

    // ---- Phase 5: write full 16 x 2048 output block (zeros outside band) ----
    // Non-temporal stores: 512 MB write-once stream; keep L2 for pop reuse.
    float* outBase = out + ((size_t)bh * T_SEQ + i0) * T_SEQ;
    const int bandStart = jt_lo * 16;
    const int bandEnd   = i0 + 16;                 // 16-aligned -> float4 homogeneous
    for (int idx = tid; idx < 16 * (T_SEQ / 4); idx += 128) {
        const int m   = idx >> 9;                  // 512 float4 per row
        const int col = (idx & 511) * 4;
        f4 v = {0.f, 0.f, 0.f, 0.f};
        if (col >= bandStart && col < bandEnd) {
            const float rs = rowScale[m];
            const float* bp = &band[m * BAND_STRIDE + (col - bandStart)];
            v.x = bp[0] * rs; v.y = bp[1] * rs; v.z = bp[2] * rs; v.w = bp[3] * rs;
        }
        __builtin_nontemporal_store(v, (f4*)(outBase + (size_t)m * T_SEQ + col));
    }
}

extern "C" void kernel_launch(void* const* d_in, const int* in_sizes, int n_in,
                              void* d_out, int out_size, void* d_ws, size_t ws_size,
                              hipStream_t stream) {
    (void)n_in; (void)out_size; (void)d_ws; (void)ws_size;
    const float* pop = (const float*)d_in[0];
    float* out = (float*)d_out;
    const int BH = in_sizes[0] / (T_SEQ * DDIM);   // B*H = 32
    dim3 grid(T_SEQ / 16, BH);
    genetic_fitness_kernel<<<grid, 128, 0, stream>>>(pop, out);
}